// BDRRAA_63136019251674
// MI455X (gfx1250) — compile-verified
//
#include <hip/hip_runtime.h>
#include <hip/hip_bf16.h>
#include <math.h>

typedef __attribute__((ext_vector_type(2))) float v2f;
typedef __attribute__((ext_vector_type(8))) float v8f;

#define K_DIM 25
#define EPSF  1e-6f
static constexpr int N_I = 100000, N_J = 50000;
static constexpr int S_I = 8192,  S_J = 4096;
static constexpr int E_CNT = 1000000;

// ---------------- zero helper (ws/accumulators are poisoned by harness) -----
__global__ void k_zero(float* __restrict__ p, int n) {
    int i = blockIdx.x * blockDim.x + threadIdx.x;
    if (i < n) p[i] = 0.f;
}

// ---------------- per-column softmax stats + colsum reduction ---------------
// Z is (K, N) row-major. G is (N, K) row-major.
// mx[c], inv[c] = softmax stats; colsum[k] += sum_c softmax(Z)[k,c]*sigmoid(G[c,k])
__global__ void __launch_bounds__(256) k_stats(const float* __restrict__ Z,
        const float* __restrict__ G, int N,
        float* __restrict__ mx, float* __restrict__ inv,
        float* __restrict__ colsum) {
    int c  = blockIdx.x * blockDim.x + threadIdx.x;
    int cc = c < N ? c : N - 1;
    float act = c < N ? 1.f : 0.f;
    float zb[K_DIM];
    float m = -3.4e38f;
    for (int k = 0; k < K_DIM; ++k) { float z = Z[k * N + cc]; zb[k] = z; m = fmaxf(m, z); }
    float s = 0.f;
    for (int k = 0; k < K_DIM; ++k) { float e = expf(zb[k] - m); zb[k] = e; s += e; }
    float is = 1.f / s;
    if (c < N) { mx[c] = m; inv[c] = is; }
    const float* g = G + (size_t)cc * K_DIM;
    int lane = threadIdx.x & 31;
    for (int k = 0; k < K_DIM; ++k) {
        float zn = zb[k] * is;
        float sg = 1.f / (1.f + expf(-g[k]));
        float v  = act * zn * sg;
        for (int o = 16; o > 0; o >>= 1) v += __shfl_xor(v, o, 32);
        if (lane == 0) atomicAdd(&colsum[k], v);
    }
}

// ---------------- ZC = Zs @ Cs (25x25, padded 32x32) via V_WMMA_F32_16X16X4 -
// P[s][m] = softmax(Z)[m, idx[s]]            (A matrix, rows m)
// Q[s][n] = P[s][n] * sigmoid(G[idx[s],n]) / colsum[n]   (B matrix)
// Block = 128 threads = 4 waves, each wave owns one 16x16 tile of the 32x32 D.
__global__ void __launch_bounds__(128) k_zc(const float* __restrict__ Z,
        const float* __restrict__ G, const int* __restrict__ idx, int S, int N,
        const float* __restrict__ mx, const float* __restrict__ inv,
        const float* __restrict__ colsum, float* __restrict__ ZC) {
    __shared__ float P[32][33];
    __shared__ float Q[32][33];
    __shared__ float csi[32];
    int t = threadIdx.x;
    int lane = t & 31;
    int w = t >> 5;
    int mbase = (w >> 1) * 16;
    int nbase = (w & 1) * 16;
    int lo = lane & 15;
    int hi = lane >> 4;
    if (t < 32) csi[t] = (t < K_DIM) ? 1.f / colsum[t] : 0.f;
    __syncthreads();
    v8f acc = {};
    int nchunk = S >> 5;
    for (int ch = blockIdx.x; ch < nchunk; ch += gridDim.x) {
        for (int r = 0; r < 8; ++r) {             // stage 32 samples x 32 rows
            int q = t + 128 * r;
            int s = q >> 5;
            int m = q & 31;
            int c = idx[(ch << 5) + s];
            float zn = 0.f, qv = 0.f;
            if (m < K_DIM) {
                zn = expf(Z[m * N + c] - mx[c]) * inv[c];
                float gg = G[c * K_DIM + m];
                qv = zn * (1.f / (1.f + expf(-gg))) * csi[m];
            }
            P[s][m] = zn;
            Q[s][m] = qv;
        }
        __syncthreads();
        for (int kk = 0; kk < 8; ++kk) {          // 8 wmma steps of K=4
            int sb = (kk << 2) + (hi << 1);
            v2f a, b;
            a.x = P[sb][mbase + lo];
            a.y = P[sb + 1][mbase + lo];
            b.x = Q[sb][nbase + lo];
            b.y = Q[sb + 1][nbase + lo];
            acc = __builtin_amdgcn_wmma_f32_16x16x4_f32(
                      false, a, false, b, (short)0, acc, false, false);
        }
        __syncthreads();
    }
    for (int r = 0; r < 8; ++r) {
        int row = mbase + r + (hi ? 8 : 0);
        int col = nbase + lo;
        atomicAdd(&ZC[row * 32 + col], acc[r]);
    }
}

// ---------------- B = A (2x25) @ ZC (25x25), stored 2x32 --------------------
__global__ void k_B(const float* __restrict__ A, const float* __restrict__ ZC,
                    float* __restrict__ B) {
    int t = threadIdx.x;
    if (t >= 64) return;
    int d = t >> 5, k = t & 31;
    float s = 0.f;
    if (k < K_DIM)
        for (int kk = 0; kk < K_DIM; ++kk) s += A[d * K_DIM + kk] * ZC[kk * 32 + k];
    B[d * 32 + k] = s;
}

// ---------------- x[s] = B @ softmaxcol(idx[s]), bias gather ----------------
__global__ void __launch_bounds__(256) k_points(const float* __restrict__ Z,
        const int* __restrict__ idx, int S, int N,
        const float* __restrict__ mx, const float* __restrict__ inv,
        const float* __restrict__ B, const float* __restrict__ biasSrc,
        float* __restrict__ X, float* __restrict__ Bias) {
    int s = blockIdx.x * blockDim.x + threadIdx.x;
    if (s >= S) return;
    int c = idx[s];
    float m = mx[c], iv = inv[c];
    float x0 = 0.f, x1 = 0.f;
    for (int k = 0; k < K_DIM; ++k) {
        float zn = expf(Z[k * N + c] - m) * iv;
        x0 += B[k] * zn;
        x1 += B[32 + k] * zn;
    }
    X[2 * s] = x0; X[2 * s + 1] = x1;
    Bias[s] = biasSrc[c];
}

// ---------------- pairwise sum: sum exp(b_p+g_q - dist), diag excluded ------
__global__ void __launch_bounds__(256) k_pair(const float* __restrict__ xi,
        const float* __restrict__ bi, const float* __restrict__ xj,
        const float* __restrict__ gj, double* __restrict__ accPair) {
    __shared__ float sx0[32], sx1[32], sb[32];
    __shared__ float red[256];
    int t = threadIdx.x;
    int ib = blockIdx.y << 5;
    if (t < 32) { int r = ib + t; sx0[t] = xi[2*r]; sx1[t] = xi[2*r+1]; sb[t] = bi[r]; }
    __syncthreads();
    int jj = (blockIdx.x << 8) + t;
    float xjx = xj[2 * jj], xjy = xj[2 * jj + 1], g = gj[jj];
    float sum = 0.f;
    for (int ii = 0; ii < 32; ++ii) {
        float dx = sx0[ii] - xjx + EPSF, dy = sx1[ii] - xjy + EPSF;
        float d  = sqrtf(dx * dx + dy * dy);
        float v  = expf(sb[ii] + g - d);
        sum += ((ib + ii) == jj) ? 0.f : v;    // M.at[q,q] = 0
    }
    red[t] = sum; __syncthreads();
    for (int o = 128; o > 0; o >>= 1) { if (t < o) red[t] += red[t + o]; __syncthreads(); }
    if (t == 0) atomicAdd(accPair, (double)red[0]);
}

// ---------------- sparse edges: sum beta_i+beta_j - ||Bi z_i - Bj z_j + eps||
__global__ void __launch_bounds__(256) k_edges(const float* __restrict__ Zi,
        const float* __restrict__ Zj, const int* __restrict__ si,
        const int* __restrict__ sj,
        const float* __restrict__ mxi, const float* __restrict__ invi,
        const float* __restrict__ mxj, const float* __restrict__ invj,
        const float* __restrict__ Bi, const float* __restrict__ Bj,
        const float* __restrict__ beta, double* __restrict__ accEdge) {
    __shared__ float sBi[64], sBj[64], red[256];
    int t = threadIdx.x;
    if (t < 64) sBi[t] = Bi[t];
    else if (t < 128) sBj[t - 64] = Bj[t - 64];
    __syncthreads();
    int e = blockIdx.x * 256 + t;
    float term = 0.f;
    if (e < E_CNT) {
        int ci = si[e], cj = sj[e];
        float mi0 = 0.f, mi1 = 0.f, mj0 = 0.f, mj1 = 0.f;
        float m = mxi[ci], iv = invi[ci];
        for (int k = 0; k < K_DIM; ++k) {
            float zn = expf(Zi[k * N_I + ci] - m) * iv;
            mi0 += sBi[k] * zn; mi1 += sBi[32 + k] * zn;
        }
        m = mxj[cj]; iv = invj[cj];
        for (int k = 0; k < K_DIM; ++k) {
            float zn = expf(Zj[k * N_J + cj] - m) * iv;
            mj0 += sBj[k] * zn; mj1 += sBj[32 + k] * zn;
        }
        float dx = mi0 - mj0 + EPSF, dy = mi1 - mj1 + EPSF;
        term = beta[ci] + beta[cj] - sqrtf(dx * dx + dy * dy);
    }
    red[t] = term; __syncthreads();
    for (int o = 128; o > 0; o >>= 1) { if (t < o) red[t] += red[t + o]; __syncthreads(); }
    if (t == 0) atomicAdd(accEdge, (double)red[0]);
}

// ---------------- final: z_pdist2 - 0.5*e^2 * pair_sum ----------------------
__global__ void k_final(const double* __restrict__ acc, float* __restrict__ out) {
    // 0.5 * e * e = 0.5 * exp(2)
    out[0] = (float)(acc[1] - 3.6945280494653247 * acc[0]);
}

extern "C" void kernel_launch(void* const* d_in, const int* in_sizes, int n_in,
                              void* d_out, int out_size, void* d_ws, size_t ws_size,
                              hipStream_t stream) {
    const float* beta  = (const float*)d_in[0];
    const float* gamma = (const float*)d_in[1];
    const float* A_i   = (const float*)d_in[2];
    const float* A_j   = (const float*)d_in[3];
    const float* Z_i   = (const float*)d_in[4];
    const float* Z_j   = (const float*)d_in[5];
    const float* G_i   = (const float*)d_in[6];
    const float* G_j   = (const float*)d_in[7];
    const int* sample_i = (const int*)d_in[8];
    const int* sample_j = (const int*)d_in[9];
    const int* sparse_i = (const int*)d_in[10];
    const int* sparse_j = (const int*)d_in[11];
    float* out = (float*)d_out;

    // workspace layout (floats) — ~1.35 MB total
    float* ws   = (float*)d_ws;
    float* maxi = ws;                  // N_I
    float* invi = maxi + N_I;          // N_I
    float* maxj = invi + N_I;          // N_J
    float* invj = maxj + N_J;          // N_J
    float* cs_i = invj + N_J;          // 32
    float* cs_j = cs_i + 32;           // 32
    float* ZC_i = cs_j + 32;           // 1024
    float* ZC_j = ZC_i + 1024;         // 1024
    float* B_i  = ZC_j + 1024;         // 64
    float* B_j  = B_i + 64;            // 64
    float* xi   = B_j + 64;            // 2*S_I
    float* bi   = xi + 2 * S_I;        // S_I
    float* xjv  = bi + S_I;            // 2*S_J
    float* gjv  = xjv + 2 * S_J;       // S_J
    double* acc = (double*)(gjv + S_J); // acc[0]=pair, acc[1]=edge (8B aligned)

    k_zero<<<9, 256, 0, stream>>>(cs_i, 64 + 2048);          // colsums + ZC tiles
    k_zero<<<1, 32, 0, stream>>>((float*)acc, 4);            // both double accums

    k_stats<<<(N_I + 255) / 256, 256, 0, stream>>>(Z_i, G_i, N_I, maxi, invi, cs_i);
    k_stats<<<(N_J + 255) / 256, 256, 0, stream>>>(Z_j, G_j, N_J, maxj, invj, cs_j);

    k_zc<<<8, 128, 0, stream>>>(Z_i, G_i, sample_i, S_I, N_I, maxi, invi, cs_i, ZC_i);
    k_zc<<<8, 128, 0, stream>>>(Z_j, G_j, sample_j, S_J, N_J, maxj, invj, cs_j, ZC_j);

    k_B<<<1, 64, 0, stream>>>(A_i, ZC_i, B_i);
    k_B<<<1, 64, 0, stream>>>(A_j, ZC_j, B_j);

    k_points<<<S_I / 256, 256, 0, stream>>>(Z_i, sample_i, S_I, N_I, maxi, invi, B_i, beta,  xi,  bi);
    k_points<<<S_J / 256, 256, 0, stream>>>(Z_j, sample_j, S_J, N_J, maxj, invj, B_j, gamma, xjv, gjv);

    k_pair<<<dim3(S_J / 256, S_I / 32), 256, 0, stream>>>(xi, bi, xjv, gjv, acc);

    k_edges<<<(E_CNT + 255) / 256, 256, 0, stream>>>(Z_i, Z_j, sparse_i, sparse_j,
                                                     maxi, invi, maxj, invj,
                                                     B_i, B_j, beta, acc + 1);

    k_final<<<1, 1, 0, stream>>>(acc, out);
}